// GIPADeepConv_15496242004293
// MI455X (gfx1250) — compile-verified
//
#include <hip/hip_runtime.h>
#include <hip/hip_bf16.h>
#include <math.h>

#define N_HEAD 8
#define NODE_F 256
#define EDGE_F 16
#define OUT_F 256
#define NEG_SLOPE 0.2f
#define BN_EPS 1e-5f

typedef float v2f __attribute__((ext_vector_type(2)));
typedef float v8f __attribute__((ext_vector_type(8)));
typedef unsigned int v4u __attribute__((ext_vector_type(4)));
typedef unsigned int v8u __attribute__((ext_vector_type(8)));

static __device__ __forceinline__ v8f wmma_f32_k4(v2f a, v2f b, v8f c) {
    // D = A(16x4 f32) x B(4x16 f32) + C(16x16 f32), exact f32 math on CDNA5
    return __builtin_amdgcn_wmma_f32_16x16x4_f32(
        /*neg_a=*/false, a, /*neg_b=*/false, b,
        /*c_mod=*/(short)0, c, /*reuse_a=*/false, /*reuse_b=*/false);
}

static __device__ __forceinline__ float leaky(float x) {
    return x >= 0.0f ? x : NEG_SLOPE * x;
}

// Monotonic float<->uint mapping so segment-max can use hardware u32 atomicMax
static __device__ __forceinline__ unsigned fmap(float f) {
    unsigned u = __float_as_uint(f);
    return (u & 0x80000000u) ? ~u : (u | 0x80000000u);
}
static __device__ __forceinline__ float funmap(unsigned u) {
    return (u & 0x80000000u) ? __uint_as_float(u ^ 0x80000000u)
                             : __uint_as_float(~u);
}
#define NEG_INF_MAP 0x007FFFFFu  // fmap(-inf)

// ---------------------------------------------------------------- init
__global__ void k_init(unsigned* m_dst, unsigned* m_src, float* s_dst,
                       float* s_src, float* msg, float* bn_acc, int n8) {
    int i = blockIdx.x * blockDim.x + threadIdx.x;
    int stride = gridDim.x * blockDim.x;
    for (; i < n8; i += stride) {
        m_dst[i] = NEG_INF_MAP;
        m_src[i] = NEG_INF_MAP;
        s_dst[i] = 0.0f;
        s_src[i] = 0.0f;
        msg[i] = 0.0f;
        if (i < 16) bn_acc[i] = 0.0f;
    }
}

// --------------------------------------------- node projections (K=256 -> 24)
__global__ __launch_bounds__(256) void k_node_proj(
    const float* __restrict__ feat, const float* __restrict__ Wp,
    const float* __restrict__ Was, const float* __restrict__ Wad,
    float* __restrict__ fc, float* __restrict__ as_, float* __restrict__ ad_,
    int n_nodes) {
    __shared__ float sWp[NODE_F * N_HEAD];
    __shared__ float sWas[NODE_F * N_HEAD];
    __shared__ float sWad[NODE_F * N_HEAD];
    for (int i = threadIdx.x; i < NODE_F * N_HEAD; i += 256) {
        sWp[i] = Wp[i]; sWas[i] = Was[i]; sWad[i] = Wad[i];
    }
    __syncthreads();
    int n = blockIdx.x * blockDim.x + threadIdx.x;
    if (n >= n_nodes) return;
    float afc[N_HEAD], aas[N_HEAD], aad[N_HEAD];
#pragma unroll
    for (int h = 0; h < N_HEAD; ++h) { afc[h] = 0.f; aas[h] = 0.f; aad[h] = 0.f; }
    const float4* frow = (const float4*)(feat + (size_t)n * NODE_F);
#pragma unroll 4
    for (int q = 0; q < NODE_F / 4; ++q) {
        float4 f = frow[q];
        float fe[4] = {f.x, f.y, f.z, f.w};
#pragma unroll
        for (int j = 0; j < 4; ++j) {
            int k = q * 4 + j;
#pragma unroll
            for (int h = 0; h < N_HEAD; ++h) {
                afc[h] = fmaf(fe[j], sWp[k * N_HEAD + h], afc[h]);
                aas[h] = fmaf(fe[j], sWas[k * N_HEAD + h], aas[h]);
                aad[h] = fmaf(fe[j], sWad[k * N_HEAD + h], aad[h]);
            }
        }
    }
#pragma unroll
    for (int h = 0; h < N_HEAD; ++h) {
        fc[(size_t)n * N_HEAD + h] = afc[h];
        as_[(size_t)n * N_HEAD + h] = aas[h];
        ad_[(size_t)n * N_HEAD + h] = aad[h];
    }
}

// ------------------------------------- edge pass 1: e + segment max (atomics)
__global__ __launch_bounds__(256) void k_edge_e_max(
    const float* __restrict__ fe, const int* __restrict__ src,
    const int* __restrict__ dst, const float* __restrict__ We,
    const float* __restrict__ as_, const float* __restrict__ ad_,
    float* __restrict__ e_out, unsigned* __restrict__ m_dst,
    unsigned* __restrict__ m_src, int n_edges) {
    __shared__ float sWe[EDGE_F * N_HEAD];
    for (int i = threadIdx.x; i < EDGE_F * N_HEAD; i += 256) sWe[i] = We[i];
    __syncthreads();
    int ei = blockIdx.x * blockDim.x + threadIdx.x;
    if (ei >= n_edges) return;
    int s = src[ei], d = dst[ei];
    float ef[EDGE_F];
    const float4* fr = (const float4*)(fe + (size_t)ei * EDGE_F);
#pragma unroll
    for (int q = 0; q < EDGE_F / 4; ++q) {
        float4 v = fr[q];
        ef[q * 4 + 0] = v.x; ef[q * 4 + 1] = v.y;
        ef[q * 4 + 2] = v.z; ef[q * 4 + 3] = v.w;
    }
    const float4* asr = (const float4*)(as_ + (size_t)s * N_HEAD);
    const float4* adr = (const float4*)(ad_ + (size_t)d * N_HEAD);
    float4 a0 = asr[0], a1 = asr[1], b0 = adr[0], b1 = adr[1];
    float asv[N_HEAD] = {a0.x, a0.y, a0.z, a0.w, a1.x, a1.y, a1.z, a1.w};
    float adv[N_HEAD] = {b0.x, b0.y, b0.z, b0.w, b1.x, b1.y, b1.z, b1.w};
    float eh[N_HEAD];
#pragma unroll
    for (int h = 0; h < N_HEAD; ++h) {
        float ae = 0.f;
#pragma unroll
        for (int k = 0; k < EDGE_F; ++k) ae = fmaf(ef[k], sWe[k * N_HEAD + h], ae);
        eh[h] = leaky(asv[h] + adv[h] + ae);
        atomicMax(&m_dst[(size_t)d * N_HEAD + h], fmap(eh[h]));
        atomicMax(&m_src[(size_t)s * N_HEAD + h], fmap(eh[h]));
    }
    float4* eo = (float4*)(e_out + (size_t)ei * N_HEAD);
    eo[0] = make_float4(eh[0], eh[1], eh[2], eh[3]);
    eo[1] = make_float4(eh[4], eh[5], eh[6], eh[7]);
}

// --------------------------------------- edge pass 2: exp-sum (L2 f32 atomics)
__global__ __launch_bounds__(256) void k_edge_sum(
    const float* __restrict__ e_in, const int* __restrict__ src,
    const int* __restrict__ dst, const unsigned* __restrict__ m_dst,
    const unsigned* __restrict__ m_src, float* __restrict__ s_dst,
    float* __restrict__ s_src, int n_edges) {
    int ei = blockIdx.x * blockDim.x + threadIdx.x;
    if (ei >= n_edges) return;
    int s = src[ei], d = dst[ei];
    const float4* er = (const float4*)(e_in + (size_t)ei * N_HEAD);
    float4 e0 = er[0], e1 = er[1];
    float eh[N_HEAD] = {e0.x, e0.y, e0.z, e0.w, e1.x, e1.y, e1.z, e1.w};
#pragma unroll
    for (int h = 0; h < N_HEAD; ++h) {
        float md = funmap(m_dst[(size_t)d * N_HEAD + h]);
        float ms = funmap(m_src[(size_t)s * N_HEAD + h]);
        atomicAdd(&s_dst[(size_t)d * N_HEAD + h], __expf(eh[h] - md));
        atomicAdd(&s_src[(size_t)s * N_HEAD + h], __expf(eh[h] - ms));
    }
}

// ------------------- edge pass 3: dual-softmax geo-mean, message scatter-add
__global__ __launch_bounds__(256) void k_edge_msg(
    const float* __restrict__ e_in, const int* __restrict__ src,
    const int* __restrict__ dst, const unsigned* __restrict__ m_dst,
    const unsigned* __restrict__ m_src, const float* __restrict__ s_dst,
    const float* __restrict__ s_src, const float* __restrict__ fc,
    float* __restrict__ msg, int n_edges) {
    int ei = blockIdx.x * blockDim.x + threadIdx.x;
    if (ei >= n_edges) return;
    int s = src[ei], d = dst[ei];
    const float4* er = (const float4*)(e_in + (size_t)ei * N_HEAD);
    float4 e0 = er[0], e1 = er[1];
    float eh[N_HEAD] = {e0.x, e0.y, e0.z, e0.w, e1.x, e1.y, e1.z, e1.w};
    const float4* fr = (const float4*)(fc + (size_t)s * N_HEAD);
    float4 f0 = fr[0], f1 = fr[1];
    float fv[N_HEAD] = {f0.x, f0.y, f0.z, f0.w, f1.x, f1.y, f1.z, f1.w};
#pragma unroll
    for (int h = 0; h < N_HEAD; ++h) {
        float md = funmap(m_dst[(size_t)d * N_HEAD + h]);
        float ms = funmap(m_src[(size_t)s * N_HEAD + h]);
        float ad = __expf(eh[h] - md) / s_dst[(size_t)d * N_HEAD + h];
        float as = __expf(eh[h] - ms) / s_src[(size_t)s * N_HEAD + h];
        float a = sqrtf(fmaxf(ad, 1e-9f) * fmaxf(as, 1e-9f));
        atomicAdd(&msg[(size_t)d * N_HEAD + h], fv[h] * a);
    }
}

// -------------------------------------------- BN stats: sum & sumsq per head
__global__ __launch_bounds__(256) void k_bn_acc(const float* __restrict__ msg,
                                                float* __restrict__ bn_acc,
                                                int n_nodes) {
    float sum[N_HEAD], sq[N_HEAD];
#pragma unroll
    for (int h = 0; h < N_HEAD; ++h) { sum[h] = 0.f; sq[h] = 0.f; }
    int i = blockIdx.x * blockDim.x + threadIdx.x;
    int stride = gridDim.x * blockDim.x;
    for (; i < n_nodes; i += stride) {
        const float4* mr = (const float4*)(msg + (size_t)i * N_HEAD);
        float4 m0 = mr[0], m1 = mr[1];
        float mv[N_HEAD] = {m0.x, m0.y, m0.z, m0.w, m1.x, m1.y, m1.z, m1.w};
#pragma unroll
        for (int h = 0; h < N_HEAD; ++h) {
            sum[h] += mv[h];
            sq[h] = fmaf(mv[h], mv[h], sq[h]);
        }
    }
#pragma unroll
    for (int h = 0; h < N_HEAD; ++h) {
#pragma unroll
        for (int off = 16; off > 0; off >>= 1) {
            sum[h] += __shfl_down(sum[h], off, 32);
            sq[h] += __shfl_down(sq[h], off, 32);
        }
    }
    if ((threadIdx.x & 31) == 0) {
#pragma unroll
        for (int h = 0; h < N_HEAD; ++h) {
            atomicAdd(&bn_acc[h], sum[h]);
            atomicAdd(&bn_acc[8 + h], sq[h]);
        }
    }
}

__global__ void k_bn_finalize(const float* __restrict__ bn_acc,
                              const float* __restrict__ gamma,
                              float* __restrict__ bn_stat, int n_nodes) {
    int h = threadIdx.x;
    if (h >= N_HEAD) return;
    float inv_n = 1.0f / (float)n_nodes;
    float mu = bn_acc[h] * inv_n;
    float var = bn_acc[8 + h] * inv_n - mu * mu;
    bn_stat[h] = mu;
    bn_stat[8 + h] = gamma[h] * rsqrtf(var + BN_EPS);
}

// ---- fused final stage: BN apply + (h@W_agg + feat@W_apply_dst + b) -> leaky
// ---- -> @W_apply + b, all via V_WMMA_F32_16X16X4_F32. 16 nodes per block.
// ---- feat tile staged by the Tensor Data Mover (TDM), which also inserts the
// ---- 4-DWORD-per-row bank-conflict padding in hardware (pad_interval/amount).
#define LDSP 260  // 256 + 4 pad: rows land on distinct LDS banks
__global__ __launch_bounds__(512) void k_fused_gemm(
    const float* __restrict__ feat, const float* __restrict__ msg,
    const float* __restrict__ bn_stat, const float* __restrict__ bn_beta,
    const float* __restrict__ W_agg, const float* __restrict__ b_agg,
    const float* __restrict__ W_apply_dst, const float* __restrict__ b_apply_dst,
    const float* __restrict__ W_apply, const float* __restrict__ b_apply,
    float* __restrict__ out, int n_nodes) {
    __shared__ float sfeat[16 * LDSP];
    __shared__ float srst[16 * LDSP];
    __shared__ float sh[16 * N_HEAD];

    const int tile = blockIdx.x;           // 16 nodes per tile
    const int wave = threadIdx.x >> 5;     // 16 waves: each owns 16 out cols
    const int lane = threadIdx.x & 31;
    const int half = lane >> 4;            // 0 -> rows 0..7 / K+0,1 ; 1 -> rows 8..15 / K+2,3
    const int l16 = lane & 15;
    const int ko = half * 2;
    const int col = wave * 16 + l16;

    // --- TDM: DMA the 16x256 f32 feat tile into LDS (wave 0 issues).
    // D# group0: count=1, lds_addr, 57-bit global addr, type=2.
    // D# group1: data_size=4B; pad_enable, pad_interval=7 (256 DW rows),
    //            pad_amount=3 (4 DW) -> LDSP=260 stride in hardware;
    //            tensor = 256 x n_nodes (OOB rows read 0), tile = 256 x 16,
    //            row stride 256 elements.
    if (threadIdx.x < 32) {
        unsigned lds_addr = (unsigned)(uintptr_t)(void*)&sfeat[0];
        unsigned long long ga = (unsigned long long)(uintptr_t)feat +
                                (unsigned long long)tile * 16ull * NODE_F * 4ull;
        v4u g0;
        g0[0] = 1u;                                   // count=1, gather off
        g0[1] = lds_addr;                             // LDS byte address
        g0[2] = (unsigned)(ga & 0xffffffffull);       // global_addr[31:0]
        g0[3] = (unsigned)((ga >> 32) & 0x01ffffffull) | (2u << 30);  // +type=2
        v8u g1;
        g1[0] = (2u << 16) | (1u << 20) | (7u << 22) | (3u << 25);
        g1[1] = ((unsigned)NODE_F & 0xffffu) << 16;   // tensor_dim0 lo16
        g1[2] = ((unsigned)NODE_F >> 16) | (((unsigned)n_nodes & 0xffffu) << 16);
        g1[3] = ((unsigned)n_nodes >> 16) | ((unsigned)NODE_F << 16);  // tile_dim0
        g1[4] = 16u;                                  // tile_dim1=16, tile_dim2=0
        g1[5] = (unsigned)NODE_F;                     // tensor_dim0_stride lo32
        g1[6] = 0u;
        g1[7] = 0u;
        asm volatile("tensor_load_to_lds %0, %1" :: "s"(g0), "s"(g1) : "memory");
        __builtin_amdgcn_s_wait_tensorcnt(0);
    }
    // batch-norm applied h tile (16x8)
    if (threadIdx.x < 16 * N_HEAD) {
        int m = threadIdx.x >> 3, k = threadIdx.x & 7;
        int node = min(tile * 16 + m, n_nodes - 1);
        sh[m * N_HEAD + k] =
            (msg[(size_t)node * N_HEAD + k] - bn_stat[k]) * bn_stat[8 + k] +
            bn_beta[k];
    }
    __syncthreads();

    v8f c;
    float binit = b_agg[col] + b_apply_dst[col];
#pragma unroll
    for (int r = 0; r < 8; ++r) c[r] = binit;

    // h @ W_agg (K=8)
#pragma unroll
    for (int kb = 0; kb < N_HEAD; kb += 4) {
        int k0 = kb + ko;
        v2f a, b;
        a.x = sh[l16 * N_HEAD + k0];
        a.y = sh[l16 * N_HEAD + k0 + 1];
        b.x = W_agg[k0 * OUT_F + col];
        b.y = W_agg[(k0 + 1) * OUT_F + col];
        c = wmma_f32_k4(a, b, c);
    }
    // feat @ W_apply_dst (K=256)
#pragma unroll 4
    for (int kb = 0; kb < NODE_F; kb += 4) {
        int k0 = kb + ko;
        __builtin_prefetch(&W_apply_dst[min(k0 + 32, NODE_F - 1) * OUT_F + col], 0, 1);
        v2f a, b;
        a.x = sfeat[l16 * LDSP + k0];
        a.y = sfeat[l16 * LDSP + k0 + 1];
        b.x = W_apply_dst[k0 * OUT_F + col];
        b.y = W_apply_dst[(k0 + 1) * OUT_F + col];
        c = wmma_f32_k4(a, b, c);
    }
    // leaky-relu, park intermediate in LDS for the second GEMM
#pragma unroll
    for (int r = 0; r < 8; ++r)
        srst[(r + half * 8) * LDSP + col] = leaky(c[r]);
    __syncthreads();

    v8f c2;
    float b2 = b_apply[col];
#pragma unroll
    for (int r = 0; r < 8; ++r) c2[r] = b2;
    // rst @ W_apply (K=256)
#pragma unroll 4
    for (int kb = 0; kb < OUT_F; kb += 4) {
        int k0 = kb + ko;
        __builtin_prefetch(&W_apply[min(k0 + 32, OUT_F - 1) * OUT_F + col], 0, 1);
        v2f a, b;
        a.x = srst[l16 * LDSP + k0];
        a.y = srst[l16 * LDSP + k0 + 1];
        b.x = W_apply[k0 * OUT_F + col];
        b.y = W_apply[(k0 + 1) * OUT_F + col];
        c2 = wmma_f32_k4(a, b, c2);
    }
#pragma unroll
    for (int r = 0; r < 8; ++r) {
        int node = tile * 16 + r + half * 8;
        if (node < n_nodes) out[(size_t)node * OUT_F + col] = c2[r];
    }
}

// ----------------------------------------------------------------- launcher
extern "C" void kernel_launch(void* const* d_in, const int* in_sizes, int n_in,
                              void* d_out, int out_size, void* d_ws, size_t ws_size,
                              hipStream_t stream) {
    const float* feat_src = (const float*)d_in[0];
    const float* feat_edge = (const float*)d_in[1];
    const int* src = (const int*)d_in[2];
    const int* dst = (const int*)d_in[3];
    const float* W_prop_src = (const float*)d_in[4];
    const float* W_attn_src = (const float*)d_in[5];
    const float* W_attn_dst = (const float*)d_in[6];
    const float* W_attn_edge = (const float*)d_in[7];
    const float* bn_gamma = (const float*)d_in[8];
    const float* bn_beta = (const float*)d_in[9];
    const float* W_agg = (const float*)d_in[10];
    const float* b_agg = (const float*)d_in[11];
    const float* W_apply_dst = (const float*)d_in[12];
    const float* b_apply_dst = (const float*)d_in[13];
    const float* W_apply = (const float*)d_in[14];
    const float* b_apply = (const float*)d_in[15];
    float* out = (float*)d_out;

    const int N = in_sizes[0] / NODE_F;
    const int E = in_sizes[2];

    // workspace carve (256B aligned)
    char* w = (char*)d_ws;
    auto carve = [&](size_t bytes) -> void* {
        void* p = (void*)w;
        w += (bytes + 255) & ~(size_t)255;
        return p;
    };
    float* fc = (float*)carve((size_t)N * N_HEAD * 4);
    float* as_ = (float*)carve((size_t)N * N_HEAD * 4);
    float* ad_ = (float*)carve((size_t)N * N_HEAD * 4);
    unsigned* m_dst = (unsigned*)carve((size_t)N * N_HEAD * 4);
    unsigned* m_src = (unsigned*)carve((size_t)N * N_HEAD * 4);
    float* s_dst = (float*)carve((size_t)N * N_HEAD * 4);
    float* s_src = (float*)carve((size_t)N * N_HEAD * 4);
    float* msg = (float*)carve((size_t)N * N_HEAD * 4);
    float* bn_acc = (float*)carve(16 * 4);
    float* bn_stat = (float*)carve(16 * 4);
    float* e_buf = (float*)carve((size_t)E * N_HEAD * 4);

    const int n8 = N * N_HEAD;
    k_init<<<512, 256, 0, stream>>>(m_dst, m_src, s_dst, s_src, msg, bn_acc, n8);
    k_node_proj<<<(N + 255) / 256, 256, 0, stream>>>(
        feat_src, W_prop_src, W_attn_src, W_attn_dst, fc, as_, ad_, N);
    k_edge_e_max<<<(E + 255) / 256, 256, 0, stream>>>(
        feat_edge, src, dst, W_attn_edge, as_, ad_, e_buf, m_dst, m_src, E);
    k_edge_sum<<<(E + 255) / 256, 256, 0, stream>>>(
        e_buf, src, dst, m_dst, m_src, s_dst, s_src, E);
    k_edge_msg<<<(E + 255) / 256, 256, 0, stream>>>(
        e_buf, src, dst, m_dst, m_src, s_dst, s_src, fc, msg, E);
    k_bn_acc<<<512, 256, 0, stream>>>(msg, bn_acc, N);
    k_bn_finalize<<<1, 32, 0, stream>>>(bn_acc, bn_gamma, bn_stat, N);
    k_fused_gemm<<<(N + 15) / 16, 512, 0, stream>>>(
        feat_src, msg, bn_stat, bn_beta, W_agg, b_agg, W_apply_dst, b_apply_dst,
        W_apply, b_apply, out, N);
}